// ICON_76871324663954
// MI455X (gfx1250) — compile-verified
//
#include <hip/hip_runtime.h>
#include <hip/hip_bf16.h>
#include <math.h>

// ---------------- problem constants ----------------
#define BATCH 128
#define TLEN  512
#define FDIM  300
#define EDIM  6
#define CDIM  6
#define G3    (3 * EDIM)        // 18 gate rows
#define BT    (BATCH * TLEN)    // 65536 rows of the big GEMM
#define KQ    (FDIM / 4)        // 75 k-quads

typedef __attribute__((ext_vector_type(2))) float v2f;
typedef __attribute__((ext_vector_type(8))) float v8f;

__device__ __forceinline__ float sigmf(float x) { return 1.0f / (1.0f + expf(-x)); }

// =====================================================================
// Kernel 1: gi[b*T+t, 0:18] = hist[b,t,:] @ wih.T   (bias added later)
// M = 65536, K = 300, N = 18 (padded to 32). f32 WMMA 16x16x4.
// grid = (BT/128, 2), block = 256 (8 waves, 16 M-rows per wave).
// B-matrix is pre-swizzled in LDS into the exact per-lane WMMA fragment
// layout (float2 per (k-quad, n-tile, lane)), so the inner loop is just
// global_load_b64 (A) + 2x ds_load_b64 (B) + 2x v_wmma, no repack movs.
// =====================================================================
__global__ void __launch_bounds__(256)
gi_gemm_kernel(const float* __restrict__ own_hist,
               const float* __restrict__ oth_hist,
               const float* __restrict__ own_wih,
               const float* __restrict__ oth_wih,
               float* __restrict__ gi_own,
               float* __restrict__ gi_oth) {
    // Bfrag[kq][tile][lane] = { Wt[kq*4 + half][n], Wt[kq*4 + 2 + half][n] }
    //   with n = (lane&15) + 16*tile, half = lane>>4, Wt[k][n] = W[n][k] (0-pad n>=18)
    __shared__ v2f Bfrag[KQ * 2 * 32];   // 75*64 float2 = 38.4 KB

    const int hist = blockIdx.y;
    const float* __restrict__ X  = hist ? oth_hist : own_hist;
    const float* __restrict__ W  = hist ? oth_wih  : own_wih;
    float*       __restrict__ GI = hist ? gi_oth   : gi_own;

    for (int idx = threadIdx.x; idx < KQ * 2 * 32; idx += blockDim.x) {
        const int kq = idx >> 6;
        const int tl = (idx >> 5) & 1;
        const int ln = idx & 31;
        const int n  = (ln & 15) + 16 * tl;
        const int k  = kq * 4 + (ln >> 4);
        v2f f;
        f.x = (n < G3) ? W[n * FDIM + k]     : 0.0f;
        f.y = (n < G3) ? W[n * FDIM + k + 2] : 0.0f;
        Bfrag[idx] = f;
    }
    __syncthreads();

    const int wave   = threadIdx.x >> 5;          // 0..7
    const int lane   = threadIdx.x & 31;
    const int m_base = blockIdx.x * 128 + wave * 16;
    const int a_row  = lane & 15;                 // M within tile
    const int a_k    = (lane >> 4) << 1;          // K offset 0 or 2 (half-wave)

    const float* __restrict__ Xrow = X + (size_t)(m_base + a_row) * FDIM + a_k;
    const v2f*   __restrict__ Bl   = Bfrag + lane;

    v8f acc0 = {};
    v8f acc1 = {};

    for (int kq = 0; kq < KQ; ++kq) {
        v2f a  = *(const v2f*)(Xrow + kq * 4);
        v2f b0 = Bl[kq * 64];
        v2f b1 = Bl[kq * 64 + 32];
        acc0 = __builtin_amdgcn_wmma_f32_16x16x4_f32(false, a, false, b0,
                                                     (short)0, acc0, false, false);
        acc1 = __builtin_amdgcn_wmma_f32_16x16x4_f32(false, a, false, b1,
                                                     (short)0, acc1, false, false);
    }

    // D layout: VGPR r -> M = r + 8*(lane>>4), N = lane&15 (per 16-wide tile)
    const int n     = lane & 15;
    const int m_off = (lane >> 4) * 8;
#pragma unroll
    for (int r = 0; r < 8; ++r) {
        const int row = m_base + m_off + r;
        if (n < G3)
            GI[(size_t)row * G3 + n] = acc0[r];
        const int n1 = 16 + n;
        if (n1 < G3)
            GI[(size_t)row * G3 + n1] = acc1[r];
    }
}

// =====================================================================
// Kernel 2: masked local GRU scan. grid.x = 2 (own / other),
// block = 128 threads, thread b owns batch row b, h[6] in registers.
// =====================================================================
__global__ void __launch_bounds__(128)
masked_scan_kernel(const float* __restrict__ gi_own,
                   const float* __restrict__ gi_oth,
                   const unsigned char* __restrict__ m_own,
                   const unsigned char* __restrict__ m_oth,
                   const float* __restrict__ own_whh,
                   const float* __restrict__ own_bih,
                   const float* __restrict__ own_bhh,
                   const float* __restrict__ oth_whh,
                   const float* __restrict__ oth_bih,
                   const float* __restrict__ oth_bhh,
                   float* __restrict__ own_out,
                   float* __restrict__ oth_out) {
    const int hist = blockIdx.x;
    const float* __restrict__ gi   = hist ? gi_oth : gi_own;
    const unsigned char* __restrict__ mask = hist ? m_oth : m_own;
    const float* __restrict__ whh = hist ? oth_whh : own_whh;
    const float* __restrict__ bih = hist ? oth_bih : own_bih;
    const float* __restrict__ bhh = hist ? oth_bhh : own_bhh;
    float* __restrict__ out = hist ? oth_out : own_out;

    __shared__ float Wh[G3 * EDIM];
    __shared__ float Bi[G3];
    __shared__ float Bh[G3];
    if (threadIdx.x < G3 * EDIM) Wh[threadIdx.x] = whh[threadIdx.x];
    if (threadIdx.x < G3) { Bi[threadIdx.x] = bih[threadIdx.x]; Bh[threadIdx.x] = bhh[threadIdx.x]; }
    __syncthreads();

    const int b = threadIdx.x;  // 0..127
    float h[EDIM];
#pragma unroll
    for (int e = 0; e < EDIM; ++e) h[e] = 0.0f;

    for (int t = 0; t < TLEN; ++t) {
        const float* __restrict__ g = gi + ((size_t)b * TLEN + t) * G3;
        float gg[G3], gh[G3];
#pragma unroll
        for (int j = 0; j < G3; ++j) gg[j] = g[j] + Bi[j];
#pragma unroll
        for (int j = 0; j < G3; ++j) {
            float s = Bh[j];
#pragma unroll
            for (int e = 0; e < EDIM; ++e) s = fmaf(Wh[j * EDIM + e], h[e], s);
            gh[j] = s;
        }
        const bool m = (mask[b * TLEN + t] != 0);
        float* __restrict__ o = out + ((size_t)b * TLEN + t) * EDIM;
#pragma unroll
        for (int e = 0; e < EDIM; ++e) {
            float r = sigmf(gg[e] + gh[e]);
            float z = sigmf(gg[EDIM + e] + gh[EDIM + e]);
            float nn = tanhf(gg[2 * EDIM + e] + r * gh[2 * EDIM + e]);
            float hn = (1.0f - z) * nn + z * h[e];
            if (m) h[e] = hn;
            o[e] = m ? h[e] : 0.0f;
        }
    }
}

// =====================================================================
// Kernel 3: q = tanh(fc), then hop_size x {GRU scan + online-softmax
// attention}, then classifier. Single block, thread b = batch row b.
// =====================================================================
__global__ void __launch_bounds__(128)
hops_kernel(const float* __restrict__ query,
            const float* __restrict__ fc_w,
            const float* __restrict__ fc_b,
            const unsigned char* __restrict__ m_own,
            const unsigned char* __restrict__ m_oth,
            const float* __restrict__ own_out,
            const float* __restrict__ oth_out,
            const float* __restrict__ glob_wih, const float* __restrict__ glob_whh,
            const float* __restrict__ glob_bih, const float* __restrict__ glob_bhh,
            const float* __restrict__ mem_wih,  const float* __restrict__ mem_whh,
            const float* __restrict__ mem_bih,  const float* __restrict__ mem_bhh,
            const float* __restrict__ cls_w,    const float* __restrict__ cls_b,
            const int* __restrict__ hops_ptr,
            float* __restrict__ rnn,
            float* __restrict__ outp) {
    __shared__ float Wi[G3 * EDIM];
    __shared__ float Wh[G3 * EDIM];
    __shared__ float Bi[G3];
    __shared__ float Bh[G3];

    const int b = threadIdx.x;  // 0..127

    // q = tanh(query @ fc_w.T + fc_b)
    float q[EDIM];
    {
        const float* __restrict__ x = query + (size_t)b * FDIM;
#pragma unroll
        for (int e = 0; e < EDIM; ++e) {
            const float* __restrict__ w = fc_w + (size_t)e * FDIM;
            float s = fc_b[e];
            for (int f = 0; f < FDIM; ++f) s = fmaf(w[f], x[f], s);
            q[e] = tanhf(s);
        }
    }

    const int hops = *hops_ptr;
    for (int hop = 0; hop < hops; ++hop) {
        const float* wih = hop ? mem_wih : glob_wih;
        const float* whh = hop ? mem_whh : glob_whh;
        const float* bih = hop ? mem_bih : glob_bih;
        const float* bhh = hop ? mem_bhh : glob_bhh;
        __syncthreads();
        if (threadIdx.x < G3 * EDIM) { Wi[threadIdx.x] = wih[threadIdx.x]; Wh[threadIdx.x] = whh[threadIdx.x]; }
        if (threadIdx.x < G3) { Bi[threadIdx.x] = bih[threadIdx.x]; Bh[threadIdx.x] = bhh[threadIdx.x]; }
        __syncthreads();

        float h[EDIM];
#pragma unroll
        for (int e = 0; e < EDIM; ++e) h[e] = 0.0f;

        // online softmax state
        float m_run = -3.0e38f, l_run = 0.0f;
        float wacc[EDIM];
#pragma unroll
        for (int e = 0; e < EDIM; ++e) wacc[e] = 0.0f;

        for (int t = 0; t < TLEN; ++t) {
            const size_t bt = (size_t)b * TLEN + t;
            const bool m = (m_own[bt] | m_oth[bt]) != 0;
            float x[EDIM];
            if (hop == 0) {
#pragma unroll
                for (int e = 0; e < EDIM; ++e)
                    x[e] = tanhf(own_out[bt * EDIM + e] + oth_out[bt * EDIM + e]);
            } else {
#pragma unroll
                for (int e = 0; e < EDIM; ++e) x[e] = rnn[bt * EDIM + e];
            }
            float gg[G3], gh[G3];
#pragma unroll
            for (int j = 0; j < G3; ++j) {
                float si = Bi[j], sh = Bh[j];
#pragma unroll
                for (int e = 0; e < EDIM; ++e) {
                    si = fmaf(Wi[j * EDIM + e], x[e], si);
                    sh = fmaf(Wh[j * EDIM + e], h[e], sh);
                }
                gg[j] = si; gh[j] = sh;
            }
            float ro[EDIM];
            float sdot = 0.0f;
#pragma unroll
            for (int e = 0; e < EDIM; ++e) {
                float r = sigmf(gg[e] + gh[e]);
                float z = sigmf(gg[EDIM + e] + gh[EDIM + e]);
                float nn = tanhf(gg[2 * EDIM + e] + r * gh[2 * EDIM + e]);
                h[e] = (1.0f - z) * nn + z * h[e];       // unmasked recurrence
                ro[e] = m ? h[e] : 0.0f;                 // masked output
                rnn[bt * EDIM + e] = ro[e];              // input to next hop
                sdot = fmaf(q[e], ro[e], sdot);
            }
            float s = tanhf(sdot);
            if (!m) s = -1.0e9f;
            // online softmax update
            const float mn = fmaxf(m_run, s);
            const float scale = expf(m_run - mn);
            const float p = expf(s - mn);
            l_run = l_run * scale + p;
#pragma unroll
            for (int e = 0; e < EDIM; ++e) wacc[e] = wacc[e] * scale + p * ro[e];
            m_run = mn;
        }
#pragma unroll
        for (int e = 0; e < EDIM; ++e) q[e] = tanhf(q[e] + wacc[e] / l_run);
    }

    // classifier: out = q @ cls_w.T + cls_b
#pragma unroll
    for (int c = 0; c < CDIM; ++c) {
        float s = cls_b[c];
#pragma unroll
        for (int e = 0; e < EDIM; ++e) s = fmaf(q[e], cls_w[c * EDIM + e], s);
        outp[(size_t)b * CDIM + c] = s;
    }
}

// =====================================================================
// Launch
// =====================================================================
extern "C" void kernel_launch(void* const* d_in, const int* in_sizes, int n_in,
                              void* d_out, int out_size, void* d_ws, size_t ws_size,
                              hipStream_t stream) {
    const float* query     = (const float*)d_in[0];
    const float* own_hist  = (const float*)d_in[1];
    const float* oth_hist  = (const float*)d_in[2];
    const unsigned char* m_own = (const unsigned char*)d_in[3];  // jax bool = 1 byte
    const unsigned char* m_oth = (const unsigned char*)d_in[4];
    const float* fc_w      = (const float*)d_in[5];
    const float* fc_b      = (const float*)d_in[6];
    const float* own_wih   = (const float*)d_in[7];
    const float* own_whh   = (const float*)d_in[8];
    const float* own_bih   = (const float*)d_in[9];
    const float* own_bhh   = (const float*)d_in[10];
    const float* oth_wih   = (const float*)d_in[11];
    const float* oth_whh   = (const float*)d_in[12];
    const float* oth_bih   = (const float*)d_in[13];
    const float* oth_bhh   = (const float*)d_in[14];
    const float* glob_wih  = (const float*)d_in[15];
    const float* glob_whh  = (const float*)d_in[16];
    const float* glob_bih  = (const float*)d_in[17];
    const float* glob_bhh  = (const float*)d_in[18];
    const float* mem_wih   = (const float*)d_in[19];
    const float* mem_whh   = (const float*)d_in[20];
    const float* mem_bih   = (const float*)d_in[21];
    const float* mem_bhh   = (const float*)d_in[22];
    const float* cls_w     = (const float*)d_in[23];
    const float* cls_b     = (const float*)d_in[24];
    const int*   hops_ptr  = (const int*)d_in[25];

    float* out = (float*)d_out;

    // workspace layout (floats)
    float* ws      = (float*)d_ws;
    float* gi_own  = ws;                                 // BT*18
    float* gi_oth  = gi_own + (size_t)BT * G3;           // BT*18
    float* own_out = gi_oth + (size_t)BT * G3;           // B*T*6
    float* oth_out = own_out + (size_t)BATCH * TLEN * EDIM;
    float* rnn     = oth_out + (size_t)BATCH * TLEN * EDIM;

    // 1) big WMMA GEMMs: gi = hist @ wih.T (both histories)
    dim3 g1(BT / 128, 2, 1);
    gi_gemm_kernel<<<g1, 256, 0, stream>>>(own_hist, oth_hist, own_wih, oth_wih,
                                           gi_own, gi_oth);

    // 2) masked GRU scans (own + other in parallel via grid.x)
    masked_scan_kernel<<<dim3(2, 1, 1), 128, 0, stream>>>(
        gi_own, gi_oth, m_own, m_oth,
        own_whh, own_bih, own_bhh,
        oth_whh, oth_bih, oth_bhh,
        own_out, oth_out);

    // 3) hops + classifier (single block, one thread per batch row)
    hops_kernel<<<dim3(1, 1, 1), 128, 0, stream>>>(
        query, fc_w, fc_b, m_own, m_oth, own_out, oth_out,
        glob_wih, glob_whh, glob_bih, glob_bhh,
        mem_wih, mem_whh, mem_bih, mem_bhh,
        cls_w, cls_b, hops_ptr, rnn, out);
}